// PytorchHMM_8778913153336
// MI455X (gfx1250) — compile-verified
//
#include <hip/hip_runtime.h>

typedef __attribute__((ext_vector_type(2))) float v2f;
typedef __attribute__((ext_vector_type(4))) float v4f;
typedef __attribute__((ext_vector_type(8))) float v8f;

#define HMM_B 512
#define HMM_T 4096
#define HMM_H 32
#define MIN_VAR 1e-6f
#define HMM_EPS 1e-10f

#define LN2f   0.6931471805599453f
#define LOG2Ef 1.4426950408889634f

// Workspace layout (floats):
// [0,1024)    Texp[i*32+j] = softmax(base_logits)[i][j] + EPS   (== exp(trans_log))
// [1024,1056) loginit[h]   = log(softmax(init)[h] + EPS)
// [1056,1088) mean[h]
// [1088,1120) inv[h]  = -0.5/var[h]
// [1120,1152) cst[h]  = -0.5*log(2*pi*var[h])

__global__ void hmm_precompute_kernel(const float* __restrict__ tl,
                                      const float* __restrict__ il,
                                      const float* __restrict__ means,
                                      const float* __restrict__ logvars,
                                      float* __restrict__ ws) {
  const int h = threadIdx.x;  // 32 threads, one per hidden state
  float m = -INFINITY;
  for (int j = 0; j < HMM_H; ++j) m = fmaxf(m, tl[h * HMM_H + j]);
  float s = 0.f;
  for (int j = 0; j < HMM_H; ++j) s += __expf(tl[h * HMM_H + j] - m);
  const float rs = 1.f / s;
  for (int j = 0; j < HMM_H; ++j)
    ws[h * HMM_H + j] = __expf(tl[h * HMM_H + j] - m) * rs + HMM_EPS;
  float mi = -INFINITY;
  for (int j = 0; j < HMM_H; ++j) mi = fmaxf(mi, il[j]);
  float si = 0.f;
  for (int j = 0; j < HMM_H; ++j) si += __expf(il[j] - mi);
  ws[1024 + h] = __logf(__expf(il[h] - mi) / si + HMM_EPS);
  const float var = __expf(logvars[h]) + MIN_VAR;
  ws[1056 + h] = means[h];
  ws[1088 + h] = -0.5f / var;
  ws[1120 + h] = -0.5f * __logf(6.283185307179586f * var);
}

__device__ __forceinline__ v8f wmma4(v2f a, v2f b, v8f c) {
  // D(16x16,f32) += A(16x4,f32) x B(4x16,f32)
  return __builtin_amdgcn_wmma_f32_16x16x4_f32(false, a, false, b, (short)0, c,
                                               false, false);
}

// Cross-half swap (lane ^ 16) as pure VALU: v_permlanex16_b32 with identity
// lane selects.  Avoids ds_bpermute + dscnt waits on the critical path.
__device__ __forceinline__ float xor16(float x) {
  int i = __builtin_bit_cast(int, x);
  int r = __builtin_amdgcn_permlanex16(i, i, 0x76543210, 0xfedcba98, false, false);
  return __builtin_bit_cast(float, r);
}

// Raw hardware transcendentals (inputs guaranteed normal-range).
__device__ __forceinline__ float rlog2(float x) { return __builtin_amdgcn_logf(x); }
__device__ __forceinline__ float rexp2(float x) { return __builtin_amdgcn_exp2f(x); }

__global__ void __launch_bounds__(32)
hmm_forward_kernel(const float* __restrict__ obs,
                   const float* __restrict__ ws,
                   float* __restrict__ out) {
  const int lane = threadIdx.x;      // 0..31
  const int up   = lane >> 4;        // 0: lower half, 1: upper half
  const int bcol = lane & 15;        // batch column within tile
  const int b    = blockIdx.x * 16 + bcol;

  const float* Texp    = ws;
  const float* loginit = ws + 1024;
  const float* meanp   = ws + 1056;
  const float* invp    = ws + 1088;
  const float* cstp    = ws + 1120;

  // Per-lane state set (matches WMMA C/D row layout):
  //  lower lanes: q[i] = state i   (i<8),  state i+8  (i>=8)   -> {0..7,16..23}
  //  upper lanes: q[i] = state i+8 (i<8),  state i+16 (i>=8)   -> {8..15,24..31}
  const int h0 = up ? 8 : 0;    // q[0..7]  -> h0 + i
  const int h1 = up ? 24 : 16;  // q[8..15] -> h1 + (i-8)

  float e_mean[16], e_inv[16], e_cst[16], e_li[16];
#pragma unroll
  for (int i = 0; i < 8; ++i) {
    e_mean[i] = meanp[h0 + i]; e_inv[i] = invp[h0 + i];
    e_cst[i]  = cstp[h0 + i];  e_li[i]  = loginit[h0 + i];
    e_mean[8 + i] = meanp[h1 + i]; e_inv[8 + i] = invp[h1 + i];
    e_cst[8 + i]  = cstp[h1 + i];  e_li[8 + i]  = loginit[h1 + i];
  }

  // Loop-invariant A tiles with the K axis PRE-PERMUTED so that the B operand
  // of chunk kc is simply {p[2kc], p[2kc+1]} in every lane (no repacking!).
  // K-slot -> state map: slot 4kc+{0,1} (lower-lane B regs) -> lower-owned
  // states state_of(2kc..2kc+1); slot 4kc+{2,3} -> upper-owned states.
  // A element (M=j, K=slot) lives in lane (j mod 16, half by slot) and must
  // hold T'[state(slot)][j]; each lane half loads with its OWN state map.
  v2f A[2][8];
#pragma unroll
  for (int jh = 0; jh < 2; ++jh)
#pragma unroll
    for (int kc = 0; kc < 8; ++kc) {
      const int i0 = 2 * kc, i1 = 2 * kc + 1;
      const int s0 = (i0 < 8) ? (h0 + i0) : (h1 + (i0 - 8));
      const int s1 = (i1 < 8) ? (h0 + i1) : (h1 + (i1 - 8));
      const int jc = jh * 16 + bcol;
      v2f a;
      a.x = Texp[s0 * HMM_H + jc];
      a.y = Texp[s1 * HMM_H + jc];
      A[jh][kc] = a;
    }

  const float* ob = obs + (size_t)b * HMM_T;
  float* aout     = out + (size_t)b * HMM_T * HMM_H;

  float q[16], p[16], mrun;

  // Finish a step: store alpha (q), max-reduce, p = exp2((q-m)*log2e).
  auto finish = [&](int t) {
    float* base = aout + (size_t)t * HMM_H;
    v4f s0 = {q[0], q[1], q[2], q[3]};
    v4f s1 = {q[4], q[5], q[6], q[7]};
    v4f s2 = {q[8], q[9], q[10], q[11]};
    v4f s3 = {q[12], q[13], q[14], q[15]};
    __builtin_nontemporal_store(s0, (v4f*)(base + h0));
    __builtin_nontemporal_store(s1, (v4f*)(base + h0 + 4));
    __builtin_nontemporal_store(s2, (v4f*)(base + h1));
    __builtin_nontemporal_store(s3, (v4f*)(base + h1 + 4));
    float m = fmaxf(q[0], q[1]);
#pragma unroll
    for (int i = 2; i < 16; ++i) m = fmaxf(m, q[i]);
    m = fmaxf(m, xor16(m));
    mrun = m;
    const float mh = m * LOG2Ef;
#pragma unroll
    for (int i = 0; i < 16; ++i)
      p[i] = rexp2(__builtin_fmaf(q[i], LOG2Ef, -mh));
  };

  // t = 0: alpha0 = loginit + emis
  {
    const float o0 = ob[0];
#pragma unroll
    for (int i = 0; i < 16; ++i) {
      const float d = o0 - e_mean[i];
      q[i] = e_li[i] + e_cst[i] + e_inv[i] * d * d;
    }
    finish(0);
  }

  float onext = ob[1];
  for (int t = 1; t < HMM_T; ++t) {
    const float o = onext;
    onext = ob[(t + 1 < HMM_T) ? t + 1 : HMM_T - 1];   // prefetch next obs

    // S[32x16] = A x P ; B chunk kc is directly {p[2kc], p[2kc+1]}.
    // Two accumulators per M-tile halve the dependent-WMMA chain depth.
    v8f D0a = {}, D0b = {}, D1a = {}, D1b = {};
#pragma unroll
    for (int kc = 0; kc < 8; kc += 2) {
      v2f b0; b0.x = p[2 * kc];     b0.y = p[2 * kc + 1];
      v2f b1; b1.x = p[2 * kc + 2]; b1.y = p[2 * kc + 3];
      D0a = wmma4(A[0][kc],     b0, D0a);
      D0b = wmma4(A[0][kc + 1], b1, D0b);
      D1a = wmma4(A[1][kc],     b0, D1a);
      D1b = wmma4(A[1][kc + 1], b1, D1b);
    }

    const float mprev = mrun;
#pragma unroll
    for (int r = 0; r < 8; ++r) {
      const float S0 = D0a[r] + D0b[r];
      const float S1 = D1a[r] + D1b[r];
      const float d0 = o - e_mean[r];
      const float d1 = o - e_mean[8 + r];
      // q = mprev + ln(S) + cst + inv*d*d   (ln via raw log2 * ln2)
      q[r]     = __builtin_fmaf(rlog2(S0), LN2f,
                   __builtin_fmaf(e_inv[r], d0 * d0, mprev + e_cst[r]));
      q[8 + r] = __builtin_fmaf(rlog2(S1), LN2f,
                   __builtin_fmaf(e_inv[8 + r], d1 * d1, mprev + e_cst[8 + r]));
    }
    finish(t);
  }

  // log_likelihood[b] = m + ln(sum_h exp(alpha[T-1,h] - m))
  float s = 0.f;
#pragma unroll
  for (int i = 0; i < 16; ++i) s += p[i];
  s += xor16(s);
  if (lane < 16) {
    out[(size_t)HMM_B * HMM_T * HMM_H + b] =
        __builtin_fmaf(rlog2(s), LN2f, mrun);
  }
}

extern "C" void kernel_launch(void* const* d_in, const int* in_sizes, int n_in,
                              void* d_out, int out_size, void* d_ws, size_t ws_size,
                              hipStream_t stream) {
  const float* obs = (const float*)d_in[0];  // [B,T,C=1]
  const float* tl  = (const float*)d_in[1];  // [H,H]
  const float* il  = (const float*)d_in[2];  // [H]
  const float* mn  = (const float*)d_in[3];  // [H,1]
  const float* lv  = (const float*)d_in[4];  // [H,1]
  float* ws  = (float*)d_ws;
  float* out = (float*)d_out;                // alpha [B,T,H] then ll [B]

  hmm_precompute_kernel<<<1, 32, 0, stream>>>(tl, il, mn, lv, ws);
  hmm_forward_kernel<<<HMM_B / 16, 32, 0, stream>>>(obs, ws, out);
}